// PageTable_51316269252855
// MI455X (gfx1250) — compile-verified
//
#include <hip/hip_runtime.h>

#define PAGE_SIZE 128
#define CHUNK 4096   // pages staged per block (16 KB tile) in the free-list scan
#define TPB_ALLOC 256
#define EPT (CHUNK / TPB_ALLOC)   // elements per thread in alloc kernel

typedef __attribute__((ext_vector_type(4))) unsigned int v4u;
typedef __attribute__((ext_vector_type(8))) int v8i;
typedef __attribute__((ext_vector_type(4))) int v4i;

#if defined(__HIP_DEVICE_COMPILE__) && __has_builtin(__builtin_amdgcn_tensor_load_to_lds)
#define HAVE_TDM 1
#else
#define HAVE_TDM 0
#endif

// This toolchain's builtin arity (probe: clang-23 / therock-10.0 = 6 args)
#define TDM_CALL(g0, g1, g2, g3) \
  __builtin_amdgcn_tensor_load_to_lds((g0), (g1), (g2), (g3), (v8i){0,0,0,0,0,0,0,0}, 0)

// ---------------------------------------------------------------------------
// TDM: DMA one contiguous row of `nelem` int32 from global into LDS.
// D# per CDNA5 ISA (1-row tile, 4-byte elements).
// Must be reached by exactly one wave; follow with s_wait_tensorcnt + barrier.
// ---------------------------------------------------------------------------
#if HAVE_TDM
__device__ __forceinline__ void tdm_load_chunk(const int* gsrc, int* lds_dst,
                                               unsigned int nelem) {
  unsigned long long ga = (unsigned long long)(uintptr_t)gsrc;
  // flat LDS addressing keeps the LDS byte offset in addr[31:0]
  unsigned int laddr = (unsigned int)(uintptr_t)lds_dst;
  v4u g0;
  g0.x = 1u;                                              // count=1 (valid D#)
  g0.y = laddr;                                           // lds_addr [63:32]
  g0.z = (unsigned int)ga;                                // global_addr [95:64]
  g0.w = (unsigned int)((ga >> 32) & 0x01FFFFFFu)         // global_addr [120:96]
         | (2u << 30);                                    // type=2 ("image")
  v8i g1;
  g1[0] = (int)(2u << 16);                                // data_size=2 -> 4B
  g1[1] = (int)((nelem & 0xFFFFu) << 16);                 // tensor_dim0[15:0]
  g1[2] = (int)((nelem >> 16) & 0xFFFFu);                 // tensor_dim0[31:16]
  g1[3] = (int)((nelem & 0xFFFFu) << 16);                 // tile_dim0
  g1[4] = 0; g1[5] = 0; g1[6] = 0; g1[7] = 0;             // 1 row: dims/strides 0
  v4i gz = {0, 0, 0, 0};                                  // groups 2/3 unused
  TDM_CALL(g0, g1, gz, gz);
  __builtin_amdgcn_s_wait_tensorcnt(0);
}
#endif

// ---------------------------------------------------------------------------
// Block-wide inclusive scan (Hillis-Steele) over blockDim.x ints in LDS.
// ---------------------------------------------------------------------------
__device__ __forceinline__ int blk_incl_scan(int v, int* lds) {
  const int t = threadIdx.x;
  const int n = blockDim.x;
  lds[t] = v;
  __syncthreads();
  for (int off = 1; off < n; off <<= 1) {
    int add = (t >= off) ? lds[t - off] : 0;
    __syncthreads();
    lds[t] += add;
    __syncthreads();
  }
  return lds[t];
}

// ---------------------------------------------------------------------------
// 0) zero the ws counters (counts + segstart)
// ---------------------------------------------------------------------------
__global__ void k_init(int* counts, int* segstart, int MS) {
  int i = blockIdx.x * blockDim.x + threadIdx.x;
  if (i <= MS) { counts[i] = 0; segstart[i] = 0; }
}

// ---------------------------------------------------------------------------
// 1) per-seq token counts + run-start table (mapped id: negatives -> MS).
// ---------------------------------------------------------------------------
__global__ void k_count(const int* __restrict__ tok, int* counts, int* segstart,
                        int N, int MS) {
  int i = blockIdx.x * blockDim.x + threadIdx.x;
  if (i >= N) return;
  int raw = tok[i];
  int m = (raw < 0) ? MS : raw;
  if (m > MS) return;                      // segment_sum drops OOB ids
  atomicAdd(&counts[m], 1);
  int prev_m = -2;
  if (i > 0) { int pr = tok[i - 1]; prev_m = (pr < 0) ? MS : pr; }
  if (i == 0 || prev_m != m) segstart[m] = i;
}

// ---------------------------------------------------------------------------
// 2) single-block per-seq logic (blockDim.x == MS == 1024).
// ---------------------------------------------------------------------------
__global__ void __launch_bounds__(1024) k_seq(
    const int* __restrict__ seq_lens, const int* __restrict__ counts,
    int* offset_ws, int* old_needed_ws, int* upd_ws,
    int* new_lens_out, int* batch_seq_lens_out, int* cu_q_lens_out,
    int* num_seqs_out, int MS) {
  __shared__ int scan[1024];
  __shared__ int lds_upd[1024];
  __shared__ int lds_cum[1024];
  __shared__ int lds_nl[1024];

  const int s = threadIdx.x;
  int cnt = counts[s];
  int sl  = seq_lens[s];
  int cur = (sl < 0) ? 0 : sl;
  int nl  = (sl >= 0) ? (cur + cnt) : -1;
  int nneed = (nl + PAGE_SIZE - 1) / PAGE_SIZE;   // x >= -1 -> trunc == floor
  int oneed = (sl + PAGE_SIZE - 1) / PAGE_SIZE;
  int need  = nneed - oneed; if (need < 0) need = 0;
  int present = (cnt > 0) ? 1 : 0;

  lds_nl[s] = nl;

  // scan 1: compaction positions for present seqs
  int pincl = blk_incl_scan(present, scan);
  int pos = pincl - present;
  int num_seqs = scan[blockDim.x - 1];
  __syncthreads();

  lds_upd[s] = MS;
  __syncthreads();
  if (present) lds_upd[pos] = s;

  // scan 2: inclusive cumsum of counts -> cu_q_lens values
  int tincl = blk_incl_scan(cnt, scan);
  int total_tok = scan[blockDim.x - 1];
  __syncthreads();
  lds_cum[s] = total_tok;               // padded slots saturate at total
  __syncthreads();
  if (present) lds_cum[pos] = tincl;

  // scan 3: exclusive prefix of per-seq page demand
  int nincl = blk_incl_scan(need, scan);

  offset_ws[s] = nincl - need;
  if (s == blockDim.x - 1) offset_ws[blockDim.x] = nincl;  // total_need
  old_needed_ws[s] = oneed;
  new_lens_out[s]  = nl;

  int u = lds_upd[s];
  upd_ws[s] = u;
  batch_seq_lens_out[s] = (u < MS) ? lds_nl[u] : -1;
  cu_q_lens_out[s + 1] = lds_cum[s];
  if (s == 0) { cu_q_lens_out[0] = 0; *num_seqs_out = num_seqs; }
}

// ---------------------------------------------------------------------------
// 3) base copy of page_indices -> new_page_indices
// ---------------------------------------------------------------------------
__global__ void k_copy(const int* __restrict__ src, int* __restrict__ dst, int n) {
  int i = blockIdx.x * blockDim.x + threadIdx.x;
  if (i < n) dst[i] = src[i];
}

// ---------------------------------------------------------------------------
// 4) per-chunk free-page counts; chunk staged in LDS via TDM by wave 0
// ---------------------------------------------------------------------------
__global__ void __launch_bounds__(256) k_freecount(
    const int* __restrict__ owners, int* blkcnt, int MP) {
  __shared__ int lds_chunk[CHUNK];
  __shared__ int lcnt;
  int base = blockIdx.x * CHUNK;
  bool full = (base + CHUNK <= MP);
  if (threadIdx.x == 0) lcnt = 0;
#if HAVE_TDM
  if (full && threadIdx.x == 0)                 // wave 0 issues the DMA once
    tdm_load_chunk(owners + base, lds_chunk, CHUNK);
#else
  if (full) {
    for (int j = threadIdx.x; j < CHUNK; j += blockDim.x)
      lds_chunk[j] = owners[base + j];
  }
#endif
  __syncthreads();
  int c = 0;
  for (int j = threadIdx.x; j < CHUNK; j += blockDim.x) {
    int p = base + j;
    int v;
    if (full) {
      v = lds_chunk[j];
    } else {
      if (p + CHUNK < MP) __builtin_prefetch(&owners[p + CHUNK], 0, 1);
      v = (p < MP) ? owners[p] : 0;
    }
    if (p < MP && v == -1) c++;
  }
  atomicAdd(&lcnt, c);
  __syncthreads();
  if (threadIdx.x == 0) blkcnt[blockIdx.x] = lcnt;
}

// ---------------------------------------------------------------------------
// 5) exclusive scan of the per-chunk counts (single block)
// ---------------------------------------------------------------------------
__global__ void k_blkscan(const int* __restrict__ blkcnt, int* blkoff, int NBLK) {
  __shared__ int scan[1024];
  int t = threadIdx.x;
  int v = (t < NBLK) ? blkcnt[t] : 0;
  int incl = blk_incl_scan(v, scan);
  if (t < NBLK) blkoff[t] = incl - v;
}

// ---------------------------------------------------------------------------
// 6) allocation: r-th free page (global rank == argmin order) goes to seq s
//    with offset[s] <= r < offset[s]+need[s], slot old_needed[s]+(r-offset[s]).
//    Chunk staged through LDS via TDM; writes ALL of new_page_owners.
// ---------------------------------------------------------------------------
__global__ void __launch_bounds__(TPB_ALLOC) k_alloc(
    const int* __restrict__ owners, const int* __restrict__ blkoff,
    const int* __restrict__ offset_ws, const int* __restrict__ old_needed_ws,
    int* __restrict__ new_owners, int* __restrict__ new_page_indices,
    int MP, int MS, int PPS) {
  __shared__ int lds_chunk[CHUNK];
  __shared__ int scan[TPB_ALLOC];
  int cbase = blockIdx.x * CHUNK;
  bool full = (cbase + CHUNK <= MP);
#if HAVE_TDM
  if (full && threadIdx.x == 0)
    tdm_load_chunk(owners + cbase, lds_chunk, CHUNK);
#else
  if (full) {
    for (int j = threadIdx.x; j < CHUNK; j += blockDim.x)
      lds_chunk[j] = owners[cbase + j];
  }
#endif
  __syncthreads();

  int base = cbase + threadIdx.x * EPT;       // contiguous sub-range per thread
  int lbase = threadIdx.x * EPT;
  int own[EPT];
  int c = 0;
  for (int j = 0; j < EPT; j++) {
    int p = base + j;
    own[j] = full ? lds_chunk[lbase + j] : ((p < MP) ? owners[p] : 0);
    if (p < MP && own[j] == -1) c++;
  }
  int incl = blk_incl_scan(c, scan);
  int r = blkoff[blockIdx.x] + (incl - c);    // global free-rank
  int total_need = offset_ws[MS];
  for (int j = 0; j < EPT; j++) {
    int p = base + j;
    if (p >= MP) break;
    int o = own[j];
    if (o == -1) {
      if (r < total_need) {
        int lo = 0, hi = MS;                  // largest s with offset[s] <= r
        while (lo + 1 < hi) {
          int mid = (lo + hi) >> 1;
          if (offset_ws[mid] <= r) lo = mid; else hi = mid;
        }
        int pidx = old_needed_ws[lo] + (r - offset_ws[lo]);
        o = lo;
        if (pidx >= 0 && pidx < PPS)          // JAX scatter 'drop' OOB
          new_page_indices[lo * PPS + pidx] = p;
      }
      r++;
    }
    new_owners[p] = o;
  }
}

// ---------------------------------------------------------------------------
// 7) batch_page_indices gather
// ---------------------------------------------------------------------------
__global__ void k_batchpi(const int* __restrict__ upd, const int* __restrict__ npi,
                          int* __restrict__ out, int MS, int PPS) {
  int i = blockIdx.x * blockDim.x + threadIdx.x;
  if (i >= MS * PPS) return;
  int row = i / PPS, col = i - row * PPS;
  int u = upd[row];
  out[i] = (u < MS) ? npi[u * PPS + col] : -1;
}

// ---------------------------------------------------------------------------
// 8) per-token destination slots + position ids (JAX gather index clamping)
// ---------------------------------------------------------------------------
__global__ void k_tokens(const int* __restrict__ tok, const int* __restrict__ seq_lens,
                         const int* __restrict__ segstart, const int* __restrict__ npi,
                         int* __restrict__ dests, int* __restrict__ pos_ids,
                         int N, int MS, int PPS) {
  int i = blockIdx.x * blockDim.x + threadIdx.x;
  if (i >= N) return;
  int raw = tok[i];
  int m = (raw < 0) ? MS : raw;
  int mc = (m > MS) ? MS : m;
  int rel = i - segstart[mc];
  int safe = (m < MS) ? m : (MS - 1);
  int sl = seq_lens[safe];
  int cur = (sl < 0) ? 0 : sl;
  int cursor = cur + rel;
  int pageidx = cursor / PAGE_SIZE;
  if (pageidx < 0) pageidx = 0;
  if (pageidx >= PPS) pageidx = PPS - 1;
  int page = npi[safe * PPS + pageidx];
  dests[i]   = (page < 0) ? -1 : page * PAGE_SIZE + (cursor % PAGE_SIZE);
  pos_ids[i] = (raw < 0) ? -1 : sl + rel;
}

// ---------------------------------------------------------------------------
extern "C" void kernel_launch(void* const* d_in, const int* in_sizes, int n_in,
                              void* d_out, int out_size, void* d_ws, size_t ws_size,
                              hipStream_t stream) {
  const int* page_indices = (const int*)d_in[0];
  const int* page_owners  = (const int*)d_in[1];
  const int* seq_lens     = (const int*)d_in[2];
  const int* tok          = (const int*)d_in[3];

  const int MSxPPS = in_sizes[0];
  const int MP     = in_sizes[1];
  const int MS     = in_sizes[2];
  const int N      = in_sizes[3];
  const int PPS    = MSxPPS / MS;
  const int NBLK   = (MP + CHUNK - 1) / CHUNK;

  // output layout (flat, reference return order)
  int* out = (int*)d_out;
  size_t off = 0;
  int* o_npi = out + off; off += (size_t)MSxPPS;
  int* o_npo = out + off; off += (size_t)MP;
  int* o_nl  = out + off; off += (size_t)MS;
  int* o_bpi = out + off; off += (size_t)MSxPPS;
  int* o_bsl = out + off; off += (size_t)MS;
  int* o_cql = out + off; off += (size_t)(MS + 1);
  int* o_ns  = out + off; off += 1;
  int* o_dst = out + off; off += (size_t)N;
  int* o_pos = out + off;

  // workspace layout
  int* w = (int*)d_ws;
  int* counts    = w;                    // MS+1
  int* segstart  = counts   + (MS + 1);  // MS+1
  int* offset_ws = segstart + (MS + 1);  // MS+1
  int* old_need  = offset_ws + (MS + 1); // MS
  int* upd       = old_need + MS;        // MS
  int* blkcnt    = upd + MS;             // NBLK
  int* blkoff    = blkcnt + NBLK;        // NBLK

  k_init<<<(MS + 1 + 255) / 256, 256, 0, stream>>>(counts, segstart, MS);
  k_count<<<(N + 255) / 256, 256, 0, stream>>>(tok, counts, segstart, N, MS);
  k_seq<<<1, MS, 0, stream>>>(seq_lens, counts, offset_ws, old_need, upd,
                              o_nl, o_bsl, o_cql, o_ns, MS);
  k_copy<<<(MSxPPS + 255) / 256, 256, 0, stream>>>(page_indices, o_npi, MSxPPS);
  k_freecount<<<NBLK, 256, 0, stream>>>(page_owners, blkcnt, MP);
  int tb = 1; while (tb < NBLK) tb <<= 1; if (tb > 1024) tb = 1024;
  k_blkscan<<<1, tb, 0, stream>>>(blkcnt, blkoff, NBLK);
  k_alloc<<<NBLK, TPB_ALLOC, 0, stream>>>(page_owners, blkoff, offset_ws, old_need,
                                          o_npo, o_npi, MP, MS, PPS);
  k_batchpi<<<(MSxPPS + 255) / 256, 256, 0, stream>>>(upd, o_npi, o_bpi, MS, PPS);
  k_tokens<<<(N + 255) / 256, 256, 0, stream>>>(tok, seq_lens, segstart, o_npi,
                                                o_dst, o_pos, N, MS, PPS);
}